// ReAttention_5085241278489
// MI455X (gfx1250) — compile-verified
//
#include <hip/hip_runtime.h>
#include <math.h>

// ---------------------------------------------------------------------------
// ReAttention for MI455X (gfx1250): fp32 WMMA (V_WMMA_F32_16X16X4_F32) GEMMs,
// fused softmax in LDS, deterministic BN stats, in-place head mixing.
// Round 2: A/B fragment reuse across multiple tiles per wave (cuts redundant
// HBM/L2 traffic), b64 fragment loads.
// ---------------------------------------------------------------------------

typedef __attribute__((ext_vector_type(2))) float v2f;
typedef __attribute__((ext_vector_type(8))) float v8f;

#define BATCH 8
#define NTOK  1024
#define CH    768
#define NH    8
#define HD    96
#define BN_EPS 1e-5f
#define SROW  1040          // padded LDS row stride for the softmax tile

// ---- WMMA f32 16x16x4 fragment helpers (layouts per CDNA5 ISA 7.12.2) ------
// A (16x4, MxK): lanes 0-15 -> M=lane, {K0,K1}; lanes 16-31 -> M=lane-16, {K2,K3}
__device__ __forceinline__ v2f ldA(const float* A, int lda, int row0, int k0) {
  const int lane = threadIdx.x & 31;
  const float* p = A + (size_t)(row0 + (lane & 15)) * lda + k0 + ((lane >> 4) << 1);
  return *(const v2f*)p;   // 8B-aligned: lda multiples of 2, k offsets even
}
// B (4x16, KxN) row-major: lanes 0-15 -> N=lane, {K0,K1}; lanes 16-31 -> {K2,K3}
__device__ __forceinline__ v2f ldB(const float* Bm, int ldb, int k0, int col0) {
  const int lane = threadIdx.x & 31;
  const int n  = lane & 15;
  const int kk = (lane >> 4) << 1;
  v2f r;
  r.x = Bm[(size_t)(k0 + kk)     * ldb + col0 + n];
  r.y = Bm[(size_t)(k0 + kk + 1) * ldb + col0 + n];
  return r;
}
// B fragment where B[k][n] = Bt[n*ldt + k]  (i.e. multiply by Bt^T)
__device__ __forceinline__ v2f ldBT(const float* Bt, int ldt, int k0, int col0) {
  const int lane = threadIdx.x & 31;
  const float* p = Bt + (size_t)(col0 + (lane & 15)) * ldt + k0 + ((lane >> 4) << 1);
  return *(const v2f*)p;
}
// D (16x16): VGPR r -> rows {r, r+8}; lanes 0-15 cols, lanes 16-31 rows+8
__device__ __forceinline__ void stD(float* Dm, int ldd, int row0, int col0, v8f d) {
  const int lane = threadIdx.x & 31;
  const int n  = lane & 15;
  const int mh = (lane >> 4) << 3;
#pragma unroll
  for (int r = 0; r < 8; ++r)
    Dm[(size_t)(row0 + mh + r) * ldd + col0 + n] = d[r];
}
__device__ __forceinline__ v8f wmma_f32(v2f a, v2f b, v8f c) {
  return __builtin_amdgcn_wmma_f32_16x16x4_f32(false, a, false, b,
                                               (short)0, c, false, false);
}

// ---------------------------------------------------------------------------
// Kernel 1: per-token 3x3 'same' convs producing q (B,H,N,d), kT (B,H,d,N),
// v (B,H,N,d).  One block per token, image staged in LDS.
// ---------------------------------------------------------------------------
__global__ void __launch_bounds__(256)
k_conv_qkv(const float* __restrict__ x,  const float* __restrict__ wq,
           const float* __restrict__ wk, const float* __restrict__ wv,
           float* __restrict__ q, float* __restrict__ kT, float* __restrict__ v) {
  __shared__ float img[768];
  __shared__ float swq[81], swk[81], swv[81];
  const int tid = threadIdx.x;
  const int bn  = blockIdx.x;
  const int b = bn >> 10, n = bn & 1023;
#pragma unroll
  for (int i = 0; i < 3; ++i)
    img[tid + 256 * i] = x[(size_t)bn * CH + tid + 256 * i];
  if (tid < 81) { swq[tid] = wq[tid]; swk[tid] = wk[tid]; swv[tid] = wv[tid]; }
  __syncthreads();
#pragma unroll
  for (int i = 0; i < 3; ++i) {
    const int c = tid + 256 * i;
    const int o = c >> 8, pix = c & 255, ii = pix >> 4, jj = pix & 15;
    float aq = 0.f, ak = 0.f, av = 0.f;
#pragma unroll
    for (int ic = 0; ic < 3; ++ic)
#pragma unroll
      for (int di = 0; di < 3; ++di) {
        const int yi = ii + di - 1;
        if ((unsigned)yi >= 16u) continue;
#pragma unroll
        for (int dj = 0; dj < 3; ++dj) {
          const int xj = jj + dj - 1;
          if ((unsigned)xj >= 16u) continue;
          const float val  = img[ic * 256 + yi * 16 + xj];
          const int   widx = ((o * 3 + ic) * 3 + di) * 3 + dj;
          aq += val * swq[widx]; ak += val * swk[widx]; av += val * swv[widx];
        }
      }
    const int h = c / HD, d = c % HD;
    const size_t qi = ((size_t)(b * NH + h) * NTOK + n) * HD + d;
    q[qi] = aq;
    v[qi] = av;
    kT[((size_t)(b * NH + h) * HD + d) * NTOK + n] = ak;
  }
}

// ---------------------------------------------------------------------------
// Kernel 2: S = q.kT * scale (WMMA), fused row softmax in a 32x1024 LDS tile.
// Block = 256 threads (8 waves); wave w computes 2 row tiles x 8 column tiles
// (B fragments reused across both row tiles); writes softmaxed A0 into d_out.
// ---------------------------------------------------------------------------
__global__ void __launch_bounds__(256)
k_scores_softmax(const float* __restrict__ q, const float* __restrict__ kT,
                 float* __restrict__ attn) {
  __shared__ float S[32 * SROW];
  __shared__ float red[256];
  const int tid = threadIdx.x;
  const int wv  = tid >> 5;            // wave 0..7
  const int bh  = blockIdx.y;          // (b*8+h) 0..63
  const int row0 = blockIdx.x * 32;    // 32-row strip
  const float* qbh = q  + (size_t)bh * NTOK * HD;
  const float* kbh = kT + (size_t)bh * HD * NTOK;
  const float scale = rsqrtf((float)HD);

  v8f acc0[8] = {}, acc1[8] = {};
  for (int k0 = 0; k0 < HD; k0 += 4) {
    const v2f a0 = ldA(qbh, HD, row0,      k0);
    const v2f a1 = ldA(qbh, HD, row0 + 16, k0);
#pragma unroll
    for (int t = 0; t < 8; ++t) {
      const v2f bf = ldB(kbh, NTOK, k0, (wv * 8 + t) * 16);
      acc0[t] = wmma_f32(a0, bf, acc0[t]);
      acc1[t] = wmma_f32(a1, bf, acc1[t]);
    }
  }
  {  // scaled scores -> LDS (rows 0..15 from acc0, 16..31 from acc1)
    const int lane = tid & 31;
    const int n  = lane & 15;
    const int mh = (lane >> 4) << 3;
#pragma unroll
    for (int t = 0; t < 8; ++t) {
      const int col0 = (wv * 8 + t) * 16;
#pragma unroll
      for (int r = 0; r < 8; ++r) {
        S[(mh + r) * SROW + col0 + n]        = acc0[t][r] * scale;
        S[(16 + mh + r) * SROW + col0 + n]   = acc1[t][r] * scale;
      }
    }
  }
  __syncthreads();

  float* dstbase = attn + ((size_t)bh * NTOK + row0) * NTOK;
  for (int r = 0; r < 32; ++r) {
    float m = -3.402823e38f;
    for (int c = tid; c < NTOK; c += 256) m = fmaxf(m, S[r * SROW + c]);
    red[tid] = m; __syncthreads();
    for (int s = 128; s > 0; s >>= 1) {
      if (tid < s) red[tid] = fmaxf(red[tid], red[tid + s]);
      __syncthreads();
    }
    const float rowmax = red[0]; __syncthreads();
    float sum = 0.f;
    for (int c = tid; c < NTOK; c += 256) {
      const float e = __expf(S[r * SROW + c] - rowmax);
      S[r * SROW + c] = e;
      sum += e;
    }
    red[tid] = sum; __syncthreads();
    for (int s = 128; s > 0; s >>= 1) {
      if (tid < s) red[tid] += red[tid + s];
      __syncthreads();
    }
    const float inv = 1.0f / red[0]; __syncthreads();
    float* drow = dstbase + (size_t)r * NTOK;
    for (int c = tid; c < NTOK; c += 256) drow[c] = S[r * SROW + c] * inv;
  }
}

// ---------------------------------------------------------------------------
// Kernel 3: deterministic per-block partial BN stats of head-mixed attention.
// ---------------------------------------------------------------------------
#define STATBLKS 1024
__global__ void __launch_bounds__(256)
k_stats_partial(const float* __restrict__ attn, const float* __restrict__ rw,
                const float* __restrict__ rb, double* __restrict__ partial) {
  __shared__ float  W[64], bias[8];
  __shared__ double buf[256][16];
  const int tid = threadIdx.x;
  if (tid < 64) W[tid] = rw[tid];
  if (tid < 8)  bias[tid] = rb[tid];
  __syncthreads();
  double ls[8], lq[8];
#pragma unroll
  for (int g = 0; g < 8; ++g) { ls[g] = 0.0; lq[g] = 0.0; }
  const long long total = (long long)BATCH * NTOK * NTOK;
  for (long long pos = (long long)blockIdx.x * 256 + tid; pos < total;
       pos += (long long)STATBLKS * 256) {
    const long long bidx = pos >> 20;
    const long long rem  = pos & 1048575;
    const float* base = attn + (bidx << 23) + rem;   // b*8*N*N + n*N + m
    float a[8];
#pragma unroll
    for (int h = 0; h < 8; ++h) a[h] = base[(size_t)h << 20];
#pragma unroll
    for (int g = 0; g < 8; ++g) {
      float m = bias[g];
#pragma unroll
      for (int h = 0; h < 8; ++h) m += W[g * 8 + h] * a[h];
      ls[g] += (double)m;
      lq[g] += (double)m * (double)m;
    }
  }
#pragma unroll
  for (int g = 0; g < 8; ++g) { buf[tid][g] = ls[g]; buf[tid][8 + g] = lq[g]; }
  __syncthreads();
  for (int s = 128; s > 0; s >>= 1) {
    if (tid < s)
      for (int k2 = 0; k2 < 16; ++k2) buf[tid][k2] += buf[tid + s][k2];
    __syncthreads();
  }
  if (tid < 16) partial[(size_t)blockIdx.x * 16 + tid] = buf[0][tid];
}

// Kernel 4: single-block deterministic final reduce (16 values, fixed order).
__global__ void k_stats_final(const double* __restrict__ partial,
                              double* __restrict__ gstats, int nblocks) {
  const int tid = threadIdx.x;
  if (tid < 16) {
    double s = 0.0;
    for (int i = 0; i < nblocks; ++i) s += partial[(size_t)i * 16 + tid];
    gstats[tid] = s;
  }
}

// ---------------------------------------------------------------------------
// Kernel 5: in-place head mix (1x1 conv across H) + BatchNorm normalize.
// Each thread owns one (b,n,m) across all 8 heads: read all, then write all.
// ---------------------------------------------------------------------------
__global__ void __launch_bounds__(256)
k_mix_bn(float* __restrict__ attn, const float* __restrict__ rw,
         const float* __restrict__ rb, const float* __restrict__ gamma,
         const float* __restrict__ beta, const double* __restrict__ gstats) {
  __shared__ float W[64], bias[8], sc[8], sh[8];
  const int tid = threadIdx.x;
  if (tid < 64) W[tid] = rw[tid];
  if (tid < 8) {
    bias[tid] = rb[tid];
    const double cnt  = (double)BATCH * NTOK * NTOK;
    const double mean = gstats[tid] / cnt;
    const double var  = gstats[8 + tid] / cnt - mean * mean;
    const float  inv  = (float)(1.0 / sqrt(var + (double)BN_EPS));
    sc[tid] = gamma[tid] * inv;
    sh[tid] = beta[tid] - (float)mean * gamma[tid] * inv;
  }
  __syncthreads();
  const long long pos  = (long long)blockIdx.x * 256 + tid;
  const long long bidx = pos >> 20;
  const long long rem  = pos & 1048575;
  float* base = attn + (bidx << 23) + rem;
  float a[8];
#pragma unroll
  for (int h = 0; h < 8; ++h) a[h] = base[(size_t)h << 20];
  float o[8];
#pragma unroll
  for (int g = 0; g < 8; ++g) {
    float m = bias[g];
#pragma unroll
    for (int h = 0; h < 8; ++h) m += W[g * 8 + h] * a[h];
    o[g] = m * sc[g] + sh[g];
  }
#pragma unroll
  for (int g = 0; g < 8; ++g) base[(size_t)g << 20] = o[g];
}

// ---------------------------------------------------------------------------
// Kernel 6: out0 = attn . v   (per (b,h): 1024x1024 * 1024x96), WMMA.
// Block = 256 (8 waves); wave w owns 2 row tiles and all 6 column tiles, so
// the 268 MB attn tensor is streamed from HBM exactly once, and each v
// B-fragment feeds two WMMAs.
// ---------------------------------------------------------------------------
__global__ void __launch_bounds__(256)
k_attn_v(const float* __restrict__ attn, const float* __restrict__ v,
         float* __restrict__ out0) {
  const int wv = threadIdx.x >> 5;               // 0..7
  const int bh = blockIdx.y;                     // 0..63
  const int b = bh >> 3, h = bh & 7;
  const float* abh = attn + (size_t)bh * NTOK * NTOK;
  const float* vbh = v    + (size_t)bh * NTOK * HD;
  const int row0 = (blockIdx.x * 16 + wv * 2) * 16;   // wave: rows row0..row0+31
  v8f acc0[6] = {}, acc1[6] = {};
  for (int k0 = 0; k0 < NTOK; k0 += 4) {
    if ((k0 & 127) == 0 && k0 + 256 < NTOK) {
      __builtin_prefetch(&abh[(size_t)(row0 + (threadIdx.x & 15)) * NTOK + k0 + 256], 0, 1);
      __builtin_prefetch(&abh[(size_t)(row0 + 16 + (threadIdx.x & 15)) * NTOK + k0 + 256], 0, 1);
    }
    const v2f a0 = ldA(abh, NTOK, row0,      k0);
    const v2f a1 = ldA(abh, NTOK, row0 + 16, k0);
#pragma unroll
    for (int t = 0; t < 6; ++t) {
      const v2f bf = ldB(vbh, HD, k0, t * 16);
      acc0[t] = wmma_f32(a0, bf, acc0[t]);
      acc1[t] = wmma_f32(a1, bf, acc1[t]);
    }
  }
  float* obase = out0 + (size_t)b * NTOK * CH + (size_t)h * HD;
#pragma unroll
  for (int t = 0; t < 6; ++t) {
    stD(obase, CH, row0,      t * 16, acc0[t]);
    stD(obase, CH, row0 + 16, t * 16, acc1[t]);
  }
}

// ---------------------------------------------------------------------------
// Kernel 7: out = out0 @ proj_w^T + proj_b   (8192x768 * 768x768), WMMA.
// Block = 256 (8 waves); wave w owns column group w (6 tiles = 96 cols) over a
// 32-row strip: proj_w fragments reused across both row tiles.
// ---------------------------------------------------------------------------
__global__ void __launch_bounds__(256)
k_proj(const float* __restrict__ out0, const float* __restrict__ pw,
       const float* __restrict__ pb, float* __restrict__ out) {
  const int wv = threadIdx.x >> 5;       // 0..7
  const int row0 = blockIdx.x * 32;      // 0..8160
  const int colb = wv * 96;              // wave's 6 column tiles
  v8f acc0[6] = {}, acc1[6] = {};
  for (int k0 = 0; k0 < CH; k0 += 4) {
    const v2f a0 = ldA(out0, CH, row0,      k0);
    const v2f a1 = ldA(out0, CH, row0 + 16, k0);
#pragma unroll
    for (int t = 0; t < 6; ++t) {
      const v2f bf = ldBT(pw, CH, k0, colb + t * 16);   // B[k][n] = pw[n*CH+k]
      acc0[t] = wmma_f32(a0, bf, acc0[t]);
      acc1[t] = wmma_f32(a1, bf, acc1[t]);
    }
  }
  const int lane = threadIdx.x & 31;
  const int n  = lane & 15;
  const int mh = (lane >> 4) << 3;
#pragma unroll
  for (int t = 0; t < 6; ++t) {
    const int col = colb + t * 16 + n;
    const float bias = pb[col];
#pragma unroll
    for (int r = 0; r < 8; ++r) {
      out[(size_t)(row0 + mh + r) * CH + col]      = acc0[t][r] + bias;
      out[(size_t)(row0 + 16 + mh + r) * CH + col] = acc1[t][r] + bias;
    }
  }
}

// ---------------------------------------------------------------------------
extern "C" void kernel_launch(void* const* d_in, const int* in_sizes, int n_in,
                              void* d_out, int out_size, void* d_ws, size_t ws_size,
                              hipStream_t stream) {
  (void)in_sizes; (void)n_in; (void)out_size; (void)ws_size;
  const float* x     = (const float*)d_in[0];
  const float* qw    = (const float*)d_in[1];
  const float* kw    = (const float*)d_in[2];
  const float* vw    = (const float*)d_in[3];
  const float* rw    = (const float*)d_in[4];
  const float* rb    = (const float*)d_in[5];
  const float* gamma = (const float*)d_in[6];
  const float* beta  = (const float*)d_in[7];
  const float* pw    = (const float*)d_in[8];
  const float* pb    = (const float*)d_in[9];

  float* out  = (float*)d_out;                       // (B,N,C)
  float* attn = out + (size_t)BATCH * NTOK * CH;     // (B,H,N,N)

  const size_t QSZ = (size_t)BATCH * NH * NTOK * HD; // 6,291,456 floats
  float*  wsf     = (float*)d_ws;
  float*  q       = wsf;
  float*  kT      = q  + QSZ;
  float*  v       = kT + QSZ;
  float*  o0      = v  + QSZ;                        // (B,N,C)
  double* partial = (double*)(o0 + (size_t)BATCH * NTOK * CH);
  double* gstats  = partial + (size_t)STATBLKS * 16;

  k_conv_qkv      <<<dim3(BATCH * NTOK), 256, 0, stream>>>(x, qw, kw, vw, q, kT, v);
  k_scores_softmax<<<dim3(NTOK / 32, BATCH * NH), 256, 0, stream>>>(q, kT, attn);
  k_stats_partial <<<dim3(STATBLKS), 256, 0, stream>>>(attn, rw, rb, partial);
  k_stats_final   <<<dim3(1), 32, 0, stream>>>(partial, gstats, STATBLKS);
  k_mix_bn        <<<dim3((BATCH * NTOK * NTOK) / 256), 256, 0, stream>>>(
                      attn, rw, rb, gamma, beta, gstats);
  k_attn_v        <<<dim3(NTOK / 256, BATCH * NH), 256, 0, stream>>>(attn, v, o0);
  k_proj          <<<dim3((BATCH * NTOK) / 32), 256, 0, stream>>>(o0, pw, pb, out);
}